// LSTMKF_68659347194187
// MI455X (gfx1250) — compile-verified
//
#include <hip/hip_runtime.h>

#define D14   14
#define SM    17          // LDS row stride (pad to dodge bank conflicts)
#define JITTER 1e-4f
#define MIN_Q  1e-5f
#define MIN_R  1e-4f

typedef __attribute__((ext_vector_type(2))) float v2f;
typedef __attribute__((ext_vector_type(8))) float v8f;

// One 16x16x16 f32 matmul as a chain of 4 V_WMMA_F32_16X16X4_F32, C-accumulating.
// A is read row-major from LDS (optionally column-masked).
// BT=false: B read row-major from LDS.  BT=true: B := (Bm)^T, i.e. read Bm with
// the A-operand index pattern (free transpose).
template <bool BT>
__device__ __forceinline__ v8f mm_chain(const float* A, const float* Bm,
                                        const float* colmask,
                                        v8f c, int l16, int h) {
#pragma unroll
  for (int kk = 0; kk < 4; ++kk) {
    const int k0 = 4 * kk + 2 * h;   // this lane supplies K = k0, k0+1
    v2f a, b;
    float a0 = A[l16 * SM + k0];
    float a1 = A[l16 * SM + k0 + 1];
    if (colmask) { a0 *= colmask[k0]; a1 *= colmask[k0 + 1]; }
    a.x = a0; a.y = a1;
    if (BT) {
      b.x = Bm[l16 * SM + k0];
      b.y = Bm[l16 * SM + k0 + 1];
    } else {
      b.x = Bm[k0 * SM + l16];
      b.y = Bm[(k0 + 1) * SM + l16];
    }
    c = __builtin_amdgcn_wmma_f32_16x16x4_f32(false, a, false, b, (short)0, c,
                                              false, false);
  }
  return c;
}

__global__ __launch_bounds__(32) void kf_scan(
    const float* __restrict__ xh, const float* __restrict__ z,
    const float* __restrict__ m21,
    const float* __restrict__ Wf, const float* __restrict__ bf,
    const float* __restrict__ Wq, const float* __restrict__ bq,
    const float* __restrict__ Wr, const float* __restrict__ br,
    float* __restrict__ out, int T) {
  __shared__ float sWf[196], sWq[196], sWr[196];
  __shared__ float sb[48];                       // bf | bq | br (16-padded)
  __shared__ float Pm[16 * SM], Sm[16 * SM], Gm[16 * SM], Km[16 * SM];
  __shared__ float zv[16], xv[16], xpv[16], qv[16], rv[16], mv[16], iv[16];

  const int lane = threadIdx.x;
  const int b    = blockIdx.x;
  const int l16  = lane & 15;
  const int h    = lane >> 4;

  for (int i = lane; i < 196; i += 32) { sWf[i] = Wf[i]; sWq[i] = Wq[i]; sWr[i] = Wr[i]; }
  if (lane < 16) {
    sb[lane]      = (lane < D14) ? bf[lane] : 0.f;
    sb[16 + lane] = (lane < D14) ? bq[lane] : 0.f;
    sb[32 + lane] = (lane < D14) ? br[lane] : 0.f;
    zv[lane] = 0.f; xv[lane] = 0.f; xpv[lane] = 0.f; qv[lane] = 0.f;
    rv[lane] = 0.f; mv[lane] = 0.f; iv[lane] = 0.f;
  }
#pragma unroll
  for (int k = 0; k < 8; ++k) {               // P0 = I (14x14 block), pads 0
    int idx = lane * 8 + k; int r = idx >> 4, c = idx & 15;
    Pm[r * SM + c] = (r == c && r < D14) ? 1.0f : 0.0f;
  }
  __syncthreads();

  const long long base = (long long)b * T;
#pragma unroll 1
  for (int t = 0; t < T; ++t) {
    const long long tok = base + t;
    // ---- token inputs ----
    if (lane < D14) {
      zv[lane] = z[tok * D14 + lane];
      xv[lane] = xh[tok * D14 + lane];
    }
    if (lane < 7) {
      float v  = m21[tok * 21 + 3 * lane + 2];
      float mm = (v > 0.5f) ? 1.0f : 0.0f;
      mv[2 * lane] = mm; mv[2 * lane + 1] = mm;
    }
    __syncthreads();
    // ---- tokenwise nets: xp = x@Wf+bf ; r = max(z@Wr+br, MIN_R) ----
    if (lane < D14) {
      float a0 = sb[lane], a2 = sb[32 + lane];
#pragma unroll
      for (int i = 0; i < D14; ++i) {
        a0 += xv[i] * sWf[i * D14 + lane];
        a2 += zv[i] * sWr[i * D14 + lane];
      }
      xpv[lane] = a0;
      rv[lane]  = fmaxf(a2, MIN_R);
    }
    __syncthreads();
    if (lane < D14) {
      float a1 = sb[16 + lane];
#pragma unroll
      for (int i = 0; i < D14; ++i) a1 += xpv[i] * sWq[i * D14 + lane];
      qv[lane] = fmaxf(a1, MIN_Q);
    }
    __syncthreads();
    // ---- predict: P_pr = P + diag(q) ----
    if (lane < D14) Pm[lane * SM + lane] += qv[lane];
    __syncthreads();
    // ---- S = mo*sym(P_pr) + diag(m*r + (1-m) + jitter*m) ----
#pragma unroll
    for (int k = 0; k < 8; ++k) {
      int idx = lane * 8 + k; int r = idx >> 4, c = idx & 15;
      float ps = 0.5f * (Pm[r * SM + c] + Pm[c * SM + r]);
      float s  = mv[r] * mv[c] * ps;
      if (r == c && r < D14) s += mv[r] * rv[r] + (1.0f - mv[r]) + JITTER * mv[r];
      Sm[r * SM + c] = s;
    }
    __syncthreads();
    // ---- trace-scaled diagonal loading ----
    float tv = (lane < D14) ? mv[lane] * Sm[lane * SM + lane] : 0.0f;
#pragma unroll
    for (int off = 16; off >= 1; off >>= 1) tv += __shfl_xor(tv, off, 32);
    const float dload = JITTER * fmaxf(tv, 1e-6f);
#pragma unroll
    for (int k = 0; k < 8; ++k) {              // G = S + dload*I, pads = identity
      int idx = lane * 8 + k; int r = idx >> 4, c = idx & 15;
      float g = Sm[r * SM + c];
      if (r == c) g = (r < D14) ? (g + dload) : 1.0f;
      Gm[r * SM + c] = g;
    }
    __syncthreads();
    // ---- in-place Gauss-Jordan: Gm <- Gm^{-1} (SPD, no pivoting needed) ----
    for (int k = 0; k < D14; ++k) {
      const float ip = 1.0f / Gm[k * SM + k];
      if (h == 0) {                            // scale pivot row
        float val = Gm[k * SM + l16];
        Gm[k * SM + l16] = (l16 == k) ? ip : val * ip;
      }
      __syncthreads();
      if (l16 != k) {                          // eliminate: row l16, cols h*8..h*8+7
        const float f = Gm[l16 * SM + k];
#pragma unroll
        for (int j = 0; j < 8; ++j) {
          int c = h * 8 + j;
          float cur = (c == k) ? 0.0f : Gm[l16 * SM + c];
          Gm[l16 * SM + c] = cur - f * Gm[k * SM + c];
        }
      }
      __syncthreads();
    }
    // ---- K = P_ho @ S_reg^{-1}   (A = P_pr column-masked, B = Gm) ----
    v8f acc = {};
    acc = mm_chain<false>(Pm, Gm, mv, acc, l16, h);
#pragma unroll
    for (int v = 0; v < 8; ++v) Km[(v + 8 * h) * SM + l16] = acc[v];
    __syncthreads();
    // ---- innov & state update ----
    if (lane < D14) iv[lane] = mv[lane] * (zv[lane] - xpv[lane]);
    __syncthreads();
    if (lane < D14) {
      float xo = xpv[lane];
#pragma unroll
      for (int i = 0; i < D14; ++i) xo += Km[lane * SM + i] * iv[i];
      out[tok * D14 + lane] = xo;
    }
    // ---- KS = K @ S ; store negated (f32 WMMA has no A-negate) ----
    v8f ks = {};
    ks = mm_chain<false>(Km, Sm, nullptr, ks, l16, h);
#pragma unroll
    for (int v = 0; v < 8; ++v) Gm[(v + 8 * h) * SM + l16] = -ks[v];
    __syncthreads();
    // ---- P_new = (-KS) @ K^T + P_pr ----
    v8f pc;
#pragma unroll
    for (int v = 0; v < 8; ++v) pc[v] = Pm[(v + 8 * h) * SM + l16];
    pc = mm_chain<true>(Gm, Km, nullptr, pc, l16, h);
#pragma unroll
    for (int v = 0; v < 8; ++v) Sm[(v + 8 * h) * SM + l16] = pc[v];
    __syncthreads();
#pragma unroll
    for (int k = 0; k < 8; ++k) {              // P <- 0.5*(Pn + Pn^T)
      int idx = lane * 8 + k; int r = idx >> 4, c = idx & 15;
      Pm[r * SM + c] = 0.5f * (Sm[r * SM + c] + Sm[c * SM + r]);
    }
    __syncthreads();
  }
}

extern "C" void kernel_launch(void* const* d_in, const int* in_sizes, int n_in,
                              void* d_out, int out_size, void* d_ws, size_t ws_size,
                              hipStream_t stream) {
  const float* xh  = (const float*)d_in[0];
  const float* z   = (const float*)d_in[1];
  const float* m21 = (const float*)d_in[2];
  const float* Wf  = (const float*)d_in[3];
  const float* bf  = (const float*)d_in[4];
  const float* Wq  = (const float*)d_in[5];
  const float* bq  = (const float*)d_in[6];
  const float* Wr  = (const float*)d_in[7];
  const float* br  = (const float*)d_in[8];
  float* out = (float*)d_out;

  const int T  = 512;                       // reference: B = T = H = 512
  const int BT = in_sizes[1] / D14;         // B*T
  const int B  = BT / T;

  kf_scan<<<dim3(B), dim3(32), 0, stream>>>(xh, z, m21, Wf, bf, Wq, bq, Wr, br,
                                            out, T);
}